// rhgcnConv_35871566856218
// MI455X (gfx1250) — compile-verified
//
#include <hip/hip_runtime.h>
#include <hip/hip_bf16.h>

typedef __attribute__((ext_vector_type(2))) float v2f;
typedef __attribute__((ext_vector_type(8))) float v8f;

#define N_NODES   100000
#define N_PAIRS   1600000
#define EDGE_NUM  30000
#define D         64
#define NEG_SLOPE 0.2f
#define N_TILES   (N_NODES / 16)   // 6250, exact

// ---------------------------------------------------------------------------
// Kernel 1: fused 4-way GEMM out = X @ Ww[t]^T + Wb[t] using
// V_WMMA_F32_16X16X4_F32 (full f32 precision; problem is bandwidth-bound so
// there is no reason to drop to f16/bf16).
//   t == 0 : Xv[node]            = X0            (accumulator seeded here)
//   t >= 1 : Xt[t-1][node]       = result * degV[t-1][node]
// One wave handles a 16-row x 64-col tile: 16 K-steps x 4 N-groups = 64 WMMAs.
// ---------------------------------------------------------------------------
__global__ void gemm_wmma_kernel(const float* __restrict__ X,
                                 const float* __restrict__ degV,
                                 const float* __restrict__ Ww,
                                 const float* __restrict__ Wb,
                                 float* __restrict__ Xt,
                                 float* __restrict__ Xv) {
  const int wave = threadIdx.x >> 5;
  const int lane = threadIdx.x & 31;
  const int tile = blockIdx.x * 4 + wave;
  if (tile >= N_TILES) return;              // wave-uniform: EXEC stays all-1s
  const int t = blockIdx.y;                 // weight/type index 0..3

  const int lo        = lane & 15;
  const int khalf     = (lane >> 4) << 1;   // 0 or 2 (K sub-offset per lane half)
  const int node_base = tile * 16;
  const int arow      = node_base + lo;

  // A fragments for all 16 K-steps (16x4 f32 tile each): lane<16 -> K={4s,4s+1},
  // lane>=16 -> K={4s+2,4s+3}, per ISA 32-bit A-matrix layout.
  v2f a[16];
#pragma unroll
  for (int s = 0; s < 16; ++s)
    a[s] = *(const v2f*)(X + (size_t)arow * D + 4 * s + khalf);

  const float* W = Ww + (size_t)t * D * D;  // row n = weights for output n
  v8f acc[4] = {};                          // 4 N-groups of 16 outputs

#pragma unroll
  for (int s = 0; s < 16; ++s) {
#pragma unroll
    for (int g = 0; g < 4; ++g) {
      const int n = g * 16 + lo;            // B[k,n] = W[n,k]
      v2f b = *(const v2f*)(W + (size_t)n * D + 4 * s + khalf);
      acc[g] = __builtin_amdgcn_wmma_f32_16x16x4_f32(
          /*neg_a=*/false, a[s], /*neg_b=*/false, b,
          /*c_mod=*/(short)0, acc[g], /*reuse_a=*/false, /*reuse_b=*/false);
    }
  }

  // Epilogue: C/D layout -> VGPR r holds M = r (lanes 0-15) / r+8 (lanes 16-31).
  const int mhalf = (lane >> 4) << 3;       // 0 or 8
  float dv[8];
  if (t >= 1) {
#pragma unroll
    for (int r = 0; r < 8; ++r)
      dv[r] = degV[(size_t)(t - 1) * N_NODES + node_base + mhalf + r];
  }
#pragma unroll
  for (int g = 0; g < 4; ++g) {
    const int n = g * 16 + lo;
    const float bias = Wb[t * D + n];
#pragma unroll
    for (int r = 0; r < 8; ++r) {
      const int node = node_base + mhalf + r;
      float v = acc[g][r] + bias;
      if (t == 0) {
        Xv[(size_t)node * D + n] = v;                                  // X0
      } else {
        Xt[((size_t)(t - 1) * N_NODES + node) * D + n] = v * dv[r];    // Xt*degV
      }
    }
  }
}

// ---------------------------------------------------------------------------
// Kernel 0: zero segment accumulators.
// ---------------------------------------------------------------------------
__global__ void zero_kernel(float* __restrict__ seg, float* __restrict__ cnt) {
  const int i = blockIdx.x * blockDim.x + threadIdx.x;
  if (i < EDGE_NUM * D) seg[i] = 0.0f;
  if (i < EDGE_NUM)     cnt[i] = 0.0f;
}

// ---------------------------------------------------------------------------
// Kernel 2: per-pair gather + segment-sum. Each edge only consumes its own
// type's mean (Xe1[etype(e), e]), so we gather Xt[etype(edges[p])][vertex[p]]
// only. 8 threads per pair, 8 contiguous columns each (coalesced 32B chunks).
// ---------------------------------------------------------------------------
__global__ void scatter_pairs_kernel(const float* __restrict__ Xt,
                                     const int* __restrict__ vertex,
                                     const int* __restrict__ edges,
                                     float* __restrict__ seg,
                                     float* __restrict__ cnt) {
  const int tid  = blockIdx.x * blockDim.x + threadIdx.x;
  const int pair = tid >> 3;
  if (pair >= N_PAIRS) return;
  const int c0 = (tid & 7) * 8;
  const int v  = vertex[pair];
  const int e  = edges[pair];
  const int t  = (e >= 10000) + (e >= 20000);   // searchsorted(LENGTHS, e, 'right')
  const float* src = Xt + ((size_t)t * N_NODES + v) * D + c0;
  float*       dst = seg + (size_t)e * D + c0;
  const float4 x0 = *(const float4*)(src);
  const float4 x1 = *(const float4*)(src + 4);
  atomicAdd(dst + 0, x0.x); atomicAdd(dst + 1, x0.y);
  atomicAdd(dst + 2, x0.z); atomicAdd(dst + 3, x0.w);
  atomicAdd(dst + 4, x1.x); atomicAdd(dst + 5, x1.y);
  atomicAdd(dst + 6, x1.z); atomicAdd(dst + 7, x1.w);
  if ((tid & 7) == 0) atomicAdd(cnt + e, 1.0f);
}

// ---------------------------------------------------------------------------
// Kernel 3: Xe = seg / max(cnt, 1).
// ---------------------------------------------------------------------------
__global__ void edge_mean_kernel(const float* __restrict__ seg,
                                 const float* __restrict__ cnt,
                                 float* __restrict__ Xe) {
  const int i = blockIdx.x * blockDim.x + threadIdx.x;
  if (i < EDGE_NUM * D) Xe[i] = seg[i] / fmaxf(cnt[i >> 6], 1.0f);
}

// ---------------------------------------------------------------------------
// Kernel 4: Xv[vertex[p]] += Xe[edges[p]] (d_out already seeded with X0).
// ---------------------------------------------------------------------------
__global__ void scatter_edges_kernel(const float* __restrict__ Xe,
                                     const int* __restrict__ vertex,
                                     const int* __restrict__ edges,
                                     float* __restrict__ Xv) {
  const int tid  = blockIdx.x * blockDim.x + threadIdx.x;
  const int pair = tid >> 3;
  if (pair >= N_PAIRS) return;
  const int c0 = (tid & 7) * 8;
  const int v  = vertex[pair];
  const int e  = edges[pair];
  const float* src = Xe + (size_t)e * D + c0;
  float*       dst = Xv + (size_t)v * D + c0;
  const float4 x0 = *(const float4*)(src);
  const float4 x1 = *(const float4*)(src + 4);
  atomicAdd(dst + 0, x0.x); atomicAdd(dst + 1, x0.y);
  atomicAdd(dst + 2, x0.z); atomicAdd(dst + 3, x0.w);
  atomicAdd(dst + 4, x1.x); atomicAdd(dst + 5, x1.y);
  atomicAdd(dst + 6, x1.z); atomicAdd(dst + 7, x1.w);
}

// ---------------------------------------------------------------------------
// Kernel 5: in-place row L2-normalize + leaky-relu. One wave32 per row,
// 2 elements/lane, butterfly reduction across the wave.
// ---------------------------------------------------------------------------
__global__ void finalize_kernel(float* __restrict__ Xv) {
  const int wave = threadIdx.x >> 5;
  const int lane = threadIdx.x & 31;
  const int row  = blockIdx.x * 8 + wave;
  if (row >= N_NODES) return;
  float* p = Xv + (size_t)row * D + lane * 2;
  float2 x = *(const float2*)p;
  float s = x.x * x.x + x.y * x.y;
#pragma unroll
  for (int off = 16; off >= 1; off >>= 1) s += __shfl_xor(s, off, 32);
  const float rn    = sqrtf(s);
  const float scale = (rn == 0.0f) ? 0.0f : 1.0f / rn;
  float a = x.x * scale, b = x.y * scale;
  a = (a >= 0.0f) ? a : NEG_SLOPE * a;
  b = (b >= 0.0f) ? b : NEG_SLOPE * b;
  float2 r; r.x = a; r.y = b;
  *(float2*)p = r;
}

// ---------------------------------------------------------------------------
extern "C" void kernel_launch(void* const* d_in, const int* in_sizes, int n_in,
                              void* d_out, int out_size, void* d_ws, size_t ws_size,
                              hipStream_t stream) {
  const float* X      = (const float*)d_in[0];   // (100000, 64)
  const float* degV   = (const float*)d_in[1];   // (3, 100000, 1)
  const float* Ww     = (const float*)d_in[2];   // (4, 64, 64)
  const float* Wb     = (const float*)d_in[3];   // (4, 64)
  const int*   vertex = (const int*)d_in[4];     // (1600000,)
  const int*   edges  = (const int*)d_in[5];     // (1600000,)
  float* Xv = (float*)d_out;                     // (100000, 64)

  // Workspace layout (floats): Xt | seg | cnt | Xe  (~92.3 MB total)
  float* ws  = (float*)d_ws;
  float* Xt  = ws;                               // 3 * N * 64
  float* seg = Xt + (size_t)3 * N_NODES * D;     // EDGE_NUM * 64
  float* cnt = seg + (size_t)EDGE_NUM * D;       // EDGE_NUM
  float* Xe  = cnt + EDGE_NUM;                   // EDGE_NUM * 64

  // 0. zero segment accumulators
  zero_kernel<<<(EDGE_NUM * D + 255) / 256, 256, 0, stream>>>(seg, cnt);

  // 1. WMMA GEMMs: X0 -> d_out, Xt*degV -> workspace
  {
    dim3 grid((N_TILES + 3) / 4, 4);
    gemm_wmma_kernel<<<grid, 128, 0, stream>>>(X, degV, Ww, Wb, Xt, Xv);
  }

  // 2. pair gather + segment-sum (+count)
  scatter_pairs_kernel<<<(N_PAIRS * 8 + 255) / 256, 256, 0, stream>>>(
      Xt, vertex, edges, seg, cnt);

  // 3. segment mean
  edge_mean_kernel<<<(EDGE_NUM * D + 255) / 256, 256, 0, stream>>>(seg, cnt, Xe);

  // 4. edge scatter into Xv (d_out holds X0 already)
  scatter_edges_kernel<<<(N_PAIRS * 8 + 255) / 256, 256, 0, stream>>>(
      Xe, vertex, edges, Xv);

  // 5. normalize + leaky relu, in place on d_out
  finalize_kernel<<<(N_NODES + 7) / 8, 256, 0, stream>>>(Xv);
}